// GNNEncoder_28475633172951
// MI455X (gfx1250) — compile-verified
//
#include <hip/hip_runtime.h>

typedef __attribute__((ext_vector_type(2))) float v2f;
typedef __attribute__((ext_vector_type(8))) float v8f;

#define TPB 256

// ---------------- degree / normalization ----------------

__global__ void k_deg_init(float* __restrict__ deg, int n) {
  int i = blockIdx.x * blockDim.x + threadIdx.x;
  if (i < n) deg[i] = 1.0f;  // self-loop contributes 1 to every node's degree
}

__global__ void k_deg_acc(const int* __restrict__ dst, float* __restrict__ deg, int e) {
  int i = blockIdx.x * blockDim.x + threadIdx.x;
  if (i < e) atomicAdd(&deg[dst[i]], 1.0f);
}

__global__ void k_dinv(float* __restrict__ deg, int n) {
  int i = blockIdx.x * blockDim.x + threadIdx.x;
  if (i < n) deg[i] = rsqrtf(deg[i]);  // deg >= 1 always (self-loops)
}

// ---------------- layer-1 aggregation (4-wide): a = P @ x ----------------

__global__ void k_selfloop4(const float* __restrict__ x, const float* __restrict__ dinv,
                            float* __restrict__ a, int n) {
  int i = blockIdx.x * blockDim.x + threadIdx.x;
  if (i < n) {
    float s = dinv[i];
    s *= s;
    float4 xv = ((const float4*)x)[i];
    float4 r;
    r.x = xv.x * s; r.y = xv.y * s; r.z = xv.z * s; r.w = xv.w * s;
    ((float4*)a)[i] = r;
  }
}

__global__ void k_edge4(const int* __restrict__ src, const int* __restrict__ dst,
                        const float* __restrict__ x, const float* __restrict__ dinv,
                        float* __restrict__ a, int e) {
  int i = blockIdx.x * blockDim.x + threadIdx.x;
  if (i < e) {
    int s = src[i], d = dst[i];
    float nm = dinv[s] * dinv[d];
    float4 xv = ((const float4*)x)[s];
    float* ad = a + (size_t)d * 4;
    atomicAdd(ad + 0, xv.x * nm);
    atomicAdd(ad + 1, xv.y * nm);
    atomicAdd(ad + 2, xv.z * nm);
    atomicAdd(ad + 3, xv.w * nm);
  }
}

// ---------------- layer-2 aggregation (32-wide): o = P @ g ----------------

__global__ void k_selfloop32(const float* __restrict__ g, const float* __restrict__ dinv,
                             float* __restrict__ o, int n32) {
  int i = blockIdx.x * blockDim.x + threadIdx.x;
  if (i < n32) {
    float s = dinv[i >> 5];
    s *= s;
    o[i] = g[i] * s;
  }
}

__global__ void k_edge32(const int* __restrict__ src, const int* __restrict__ dst,
                         const float* __restrict__ g, const float* __restrict__ dinv,
                         float* __restrict__ o, int e) {
  int gid = blockIdx.x * blockDim.x + threadIdx.x;
  int ei = gid >> 5;        // one wave32 per edge
  int f = gid & 31;         // lane = feature
  if (ei < e) {
    int s = src[ei], d = dst[ei];
    float nm = dinv[s] * dinv[d];
    atomicAdd(&o[(size_t)d * 32 + f], g[(size_t)s * 32 + f] * nm);
  }
}

__global__ void k_bias_relu32(float* __restrict__ h, const float* __restrict__ b, int total) {
  int i = blockIdx.x * blockDim.x + threadIdx.x;
  if (i < total) h[i] = fmaxf(h[i] + b[i & 31], 0.0f);
}

// ---------------- dense GEMM via V_WMMA_F32_16X16X4_F32 ----------------
// One wave computes a 16x16 output tile. K multiple of 4, NC multiple of 16.
// A fragment (16x4 f32): lane<16 holds A[m][k],A[m][k+1]; lane>=16 holds A[m][k+2],A[m][k+3]
// B fragment (4x16 f32): lane indexes column; VGPR0/1 = rows (k+2*half, k+1+2*half)
// C/D (16x16 f32): VGPR r -> rows r (lanes 0-15) and r+8 (lanes 16-31)
template <int K, int NC, bool BIAS_RELU>
__global__ void k_gemm_wmma(const float* __restrict__ A, const float* __restrict__ B,
                            const float* __restrict__ bias, float* __restrict__ C,
                            int mtiles, int M) {
  int gid = blockIdx.x * blockDim.x + threadIdx.x;
  int wave = gid >> 5;
  int lane = threadIdx.x & 31;
  constexpr int ntiles = NC / 16;
  int tm = wave / ntiles;
  int tn = wave % ntiles;
  if (tm >= mtiles) return;  // wave-uniform: EXEC stays all-ones for WMMA

  int l15 = lane & 15;
  int half = lane >> 4;

  int rowA = tm * 16 + l15;
  if (rowA > M - 1) rowA = M - 1;  // clamp (no-op when M % 16 == 0)
  const float* Arow = A + (size_t)rowA * K;
  int coln = tn * 16 + l15;

  v8f c = {};
#pragma unroll
  for (int k = 0; k < K; k += 4) {
    int ka = k + half * 2;
    v2f a = *(const v2f*)(Arow + ka);
    v2f b;
    b.x = B[ka * NC + coln];
    b.y = B[(ka + 1) * NC + coln];
    // 8 args: (neg_a, A, neg_b, B, c_mod, C, reuse_a, reuse_b)
    c = __builtin_amdgcn_wmma_f32_16x16x4_f32(false, a, false, b, (short)0, c, false, false);
  }

  float bv = 0.0f;
  if (BIAS_RELU) bv = bias[coln];

  int row0 = tm * 16 + half * 8;  // first of this lane's 8 output rows
  float* Cbase = C + (size_t)row0 * NC + coln;

  if (tm * 16 + 16 <= M) {
    // full tile (wave-uniform condition): unguarded stores, immediate offsets
#pragma unroll
    for (int r = 0; r < 8; ++r) {
      float v = c[r];
      if (BIAS_RELU) v = fmaxf(v + bv, 0.0f);
      Cbase[(size_t)r * NC] = v;
    }
  } else {
    // partial tail tile: per-row guard
#pragma unroll
    for (int r = 0; r < 8; ++r) {
      float v = c[r];
      if (BIAS_RELU) v = fmaxf(v + bv, 0.0f);
      if (row0 + r < M) Cbase[(size_t)r * NC] = v;
    }
  }
}

// ---------------- launcher ----------------

static inline int nblk(long long t) { return (int)((t + TPB - 1) / TPB); }

extern "C" void kernel_launch(void* const* d_in, const int* in_sizes, int n_in,
                              void* d_out, int out_size, void* d_ws, size_t ws_size,
                              hipStream_t stream) {
  const float* x  = (const float*)d_in[0];  // [N,4]
  const int*   ei = (const int*)d_in[1];    // [2,E]
  const float* W1 = (const float*)d_in[2];  // [4,64]
  const float* b1 = (const float*)d_in[3];  // [64]
  const float* W2 = (const float*)d_in[4];  // [64,32]
  const float* b2 = (const float*)d_in[5];  // [32]
  const float* Wf = (const float*)d_in[6];  // [32,32]
  const float* bf = (const float*)d_in[7];  // [32]
  float* out = (float*)d_out;               // [N,32]

  const int N = in_sizes[0] / 4;
  const int E = in_sizes[1] / 2;
  const int* src = ei;
  const int* dst = ei + E;

  // workspace layout (256B aligned chunks)
  char* ws = (char*)d_ws;
  auto take = [&](size_t bytes) {
    char* p = ws;
    ws += (bytes + 255) & ~(size_t)255;
    return (float*)p;
  };
  float* dinv = take((size_t)N * sizeof(float));        // degree -> dinv in place
  float* a4   = take((size_t)N * 4 * sizeof(float));    // P @ x
  float* h1   = take((size_t)N * 64 * sizeof(float));   // relu(a4@W1 + b1)
  float* g2   = take((size_t)N * 32 * sizeof(float));   // h1 @ W2
  float* o2   = take((size_t)N * 32 * sizeof(float));   // P @ g2, then relu(+b2) in place

  const int mtiles = (N + 15) / 16;

  // degrees -> D^{-1/2}
  k_deg_init<<<nblk(N), TPB, 0, stream>>>(dinv, N);
  k_deg_acc<<<nblk(E), TPB, 0, stream>>>(dst, dinv, E);
  k_dinv<<<nblk(N), TPB, 0, stream>>>(dinv, N);

  // layer 1: a4 = P@x  ->  h1 = relu(a4@W1 + b1)
  k_selfloop4<<<nblk(N), TPB, 0, stream>>>(x, dinv, a4, N);
  k_edge4<<<nblk(E), TPB, 0, stream>>>(src, dst, x, dinv, a4, E);
  {
    long long waves = (long long)mtiles * (64 / 16);
    k_gemm_wmma<4, 64, true><<<nblk(waves * 32), TPB, 0, stream>>>(a4, W1, b1, h1, mtiles, N);
  }

  // layer 2: g2 = h1@W2  ->  o2 = P@g2  ->  h2 = relu(o2 + b2) (in place)
  {
    long long waves = (long long)mtiles * (32 / 16);
    k_gemm_wmma<64, 32, false><<<nblk(waves * 32), TPB, 0, stream>>>(h1, W2, nullptr, g2, mtiles, N);
  }
  k_selfloop32<<<nblk((long long)N * 32), TPB, 0, stream>>>(g2, dinv, o2, N * 32);
  k_edge32<<<nblk((long long)E * 32), TPB, 0, stream>>>(src, dst, g2, dinv, o2, E);
  k_bias_relu32<<<nblk((long long)N * 32), TPB, 0, stream>>>(o2, b2, N * 32);

  // final: out = relu(h2 @ Wf + bf)
  {
    long long waves = (long long)mtiles * (32 / 16);
    k_gemm_wmma<32, 32, true><<<nblk(waves * 32), TPB, 0, stream>>>(o2, Wf, bf, out, mtiles, N);
  }
}